// CRF_57741540327625
// MI455X (gfx1250) — compile-verified
//
#include <hip/hip_runtime.h>
#include <hip/hip_bf16.h>

// ---- problem constants (from reference) ----
#define BB   64          // batch
#define TT   512         // time steps
#define DD   1024        // feature dim (K)
#define NN   64          // tags (N)
#define MM   (BB * TT)   // 32768 GEMM rows

typedef __attribute__((ext_vector_type(2))) float v2f;
typedef __attribute__((ext_vector_type(8))) float v8f;

#define KCHUNK  128
#define NCHUNK  (DD / KCHUNK)      // 8
#define ASTRIDE (KCHUNK + 4)       // 132 ≡ 4 (mod 64 banks) -> conflict-free A reads

// ============================================================================
// Kernel 0: pack W (D,N) -> Wp (D/2, N) of float2 {W[2p][n], W[2p+1][n]}
// so a B-fragment K-pair is a single coalesced b64 load.
// ============================================================================
__global__ __launch_bounds__(256) void pack_w_kernel(
    const float* __restrict__ W, float2* __restrict__ Wp)
{
    const int idx = blockIdx.x * 256 + threadIdx.x;
    if (idx < (DD / 2) * NN) {
        const int p = idx / NN;
        const int n = idx % NN;
        Wp[idx] = make_float2(W[(size_t)(2 * p) * NN + n],
                              W[(size_t)(2 * p + 1) * NN + n]);
    }
}

// ============================================================================
// Kernel 1: scores = (inputs @ W + bias) * mask   via V_WMMA_F32_16X16X4_F32
// grid.x = MM/16 M-tiles; block = 128 threads = 4 waves, wave w owns N-cols
// [16w,16w+16). A tile double-buffered in LDS via async global->LDS copies.
// ============================================================================
__device__ __forceinline__ void issue_a_chunk(
    const float* __restrict__ inputs, int row0, int k0,
    float* dstbase, int wave, int lane)
{
    #pragma unroll
    for (int i = 0; i < 4; ++i) {
        const int r = wave + 4 * i;                  // row within 16-row tile
        const unsigned voff =
            (unsigned)((row0 + r) * (DD * 4) + k0 * 4 + lane * 16);
        const unsigned ldsa =
            (unsigned)(uintptr_t)(dstbase + r * ASTRIDE + lane * 4);
        asm volatile("global_load_async_to_lds_b128 %0, %1, %2 offset:0"
                     :: "v"(ldsa), "v"(voff), "s"(inputs) : "memory");
    }
}

__global__ __launch_bounds__(128) void crf_gemm_kernel(
    const float*  __restrict__ inputs,   // (B*T, D)
    const int*    __restrict__ mask,     // (B*T)
    const float2* __restrict__ Wp,       // (D/2, N) packed K-pairs
    const float*  __restrict__ bias,     // (N)
    float*        __restrict__ scores)   // (B*T, N)
{
    __shared__ __align__(16) float As[2][16 * ASTRIDE];

    const int tid   = threadIdx.x;
    const int lane  = tid & 31;
    const int wave  = tid >> 5;            // 0..3 == N-tile
    const int row0  = blockIdx.x * 16;     // first M row of this tile

    const int m     = lane & 15;           // A-fragment row / C column id
    const int khalf = lane >> 4;           // 0: K pair {0,1}; 1: K pair {2,3}
    const int n     = wave * 16 + m;       // global N column

    v8f acc = {0.f, 0.f, 0.f, 0.f, 0.f, 0.f, 0.f, 0.f};

    // prologue: async-stage chunk 0
    issue_a_chunk(inputs, row0, 0, &As[0][0], wave, lane);
    asm volatile("s_wait_asynccnt 0x0" ::: "memory");
    __syncthreads();

    for (int c = 0; c < NCHUNK; ++c) {
        const int k0 = c * KCHUNK;

        // overlap: stage chunk c+1 while computing on chunk c
        if (c + 1 < NCHUNK)
            issue_a_chunk(inputs, row0, k0 + KCHUNK, &As[(c + 1) & 1][0],
                          wave, lane);

        const float* Ab = &As[c & 1][0];
        #pragma unroll
        for (int kk = 0; kk < KCHUNK; kk += 4) {
            const int kb = kk + khalf * 2;           // this lane's K pair
            v2f a;
            a.x = Ab[m * ASTRIDE + kb];
            a.y = Ab[m * ASTRIDE + kb + 1];

            const int kp = (k0 + kk) / 2 + khalf;    // packed K-pair index
            const float2 wv = Wp[(size_t)kp * NN + n];
            v2f bf;
            bf.x = wv.x;
            bf.y = wv.y;

            acc = __builtin_amdgcn_wmma_f32_16x16x4_f32(
                false, a, false, bf, (short)0, acc, false, false);
        }

        // chunk c+1 async copies done + visible to all waves
        asm volatile("s_wait_asynccnt 0x0" ::: "memory");
        __syncthreads();
    }

    // ---- epilogue: + bias, * mask, store ----
    const float bn = bias[n];
    #pragma unroll
    for (int v = 0; v < 8; ++v) {
        const int row = row0 + v + 8 * khalf;        // C/D layout: M = v + 8*khalf
        const float mf = (float)mask[row];
        scores[(size_t)row * NN + n] = (acc[v] + bn) * mf;
    }
}

// ============================================================================
// Kernel 2: fused Viterbi + log-partition forward recurrences + backtrace.
// grid.x = B blocks; block = 256 threads: thread = (j = tid>>2, s = tid&3);
// sub-thread s reduces i in [16s, 16s+16), merged via wave32 __shfl_xor.
// ============================================================================
#define TSTRIDE 65   // padded trans stride -> conflict-free column reads

__global__ __launch_bounds__(256) void crf_forward_kernel(
    const float* __restrict__ scores,   // (B*T, N)
    const int*   __restrict__ mask,     // (B*T)
    const float* __restrict__ trans,    // (N, N)
    const float* __restrict__ startv,   // (N)
    const float* __restrict__ endv,     // (N)
    float*       __restrict__ logz,     // (B)  [workspace]
    unsigned char* __restrict__ bp_buf, // (B*T*N) backpointers [workspace]
    float*       __restrict__ tags_out) // (B*T) tags written as float
{
    __shared__ float trans_s[NN * TSTRIDE];
    __shared__ float av_s[NN];
    __shared__ float az_s[NN];

    const int b   = blockIdx.x;
    const int tid = threadIdx.x;
    const int j   = tid >> 2;
    const int s   = tid & 3;
    const size_t bT = (size_t)b * TT;

    for (int idx = tid; idx < NN * NN; idx += 256) {
        trans_s[(idx >> 6) * TSTRIDE + (idx & 63)] = trans[idx];
    }
    __syncthreads();

    float av = startv[j] + scores[bT * NN + j];
    float az = av;

    for (int t = 1; t < TT; ++t) {
        if (s == 0) { av_s[j] = av; az_s[j] = az; }
        __syncthreads();

        const float emit = scores[(bT + t) * NN + j];
        const int   msk  = mask[bT + t];

        float best = -3.4e38f; int bp = 0;
        float mx   = -3.4e38f;
        #pragma unroll
        for (int ii = 0; ii < 16; ++ii) {
            const int   i  = s * 16 + ii;
            const float tr = trans_s[i * TSTRIDE + j];
            const float sv = av_s[i] + tr;
            if (sv > best) { best = sv; bp = i; }
            const float sz = az_s[i] + tr;
            mx = fmaxf(mx, sz);
        }
        #pragma unroll
        for (int off = 1; off < 4; off <<= 1) {
            const float ov = __shfl_xor(best, off, 32);
            const int   oi = __shfl_xor(bp,   off, 32);
            if (ov > best || (ov == best && oi < bp)) { best = ov; bp = oi; }
            mx = fmaxf(mx, __shfl_xor(mx, off, 32));
        }
        float ssum = 0.f;
        #pragma unroll
        for (int ii = 0; ii < 16; ++ii) {
            const int i = s * 16 + ii;
            ssum += __expf(az_s[i] + trans_s[i * TSTRIDE + j] - mx);
        }
        #pragma unroll
        for (int off = 1; off < 4; off <<= 1) {
            ssum += __shfl_xor(ssum, off, 32);
        }

        const float nav = best + emit;
        const float naz = mx + __logf(ssum) + emit;
        const int   nbp = msk ? bp : j;
        av = msk ? nav : av;
        az = msk ? naz : az;
        if (s == 0) bp_buf[(bT + t) * NN + j] = (unsigned char)nbp;
        __syncthreads();
    }

    if (s == 0) { av_s[j] = av + endv[j]; az_s[j] = az + endv[j]; }
    __syncthreads();

    if (tid == 0) {
        float mx = az_s[0];
        for (int k = 1; k < NN; ++k) mx = fmaxf(mx, az_s[k]);
        float ssum = 0.f;
        for (int k = 0; k < NN; ++k) ssum += __expf(az_s[k] - mx);
        logz[b] = mx + __logf(ssum);

        float best = av_s[0]; int tag = 0;
        for (int k = 1; k < NN; ++k) {
            if (av_s[k] > best) { best = av_s[k]; tag = k; }
        }
        tags_out[bT + TT - 1] = (float)(tag * mask[bT + TT - 1]);
        int cur = tag;
        for (int t = TT - 1; t >= 1; --t) {
            cur = (int)bp_buf[(bT + t) * NN + cur];
            tags_out[bT + t - 1] = (float)(cur * mask[bT + t - 1]);
        }
    }
}

// ============================================================================
// Kernel 3: gold (joint) score per batch + mean loss.  1 block, 64 threads.
// ============================================================================
__global__ __launch_bounds__(64) void crf_loss_kernel(
    const float* __restrict__ scores,
    const int*   __restrict__ mask,
    const int*   __restrict__ labels,
    const float* __restrict__ trans,
    const float* __restrict__ startv,
    const float* __restrict__ endv,
    const float* __restrict__ logz,
    float*       __restrict__ loss_out)
{
    __shared__ float red[BB];
    const int b = threadIdx.x;
    const size_t bT = (size_t)b * TT;

    int prev = labels[bT];
    float g = startv[prev];
    int lastidx = 0;
    for (int t = 0; t < TT; ++t) {
        const int   lab = labels[bT + t];
        const int   mi  = mask[bT + t];
        const float mf  = (float)mi;
        if (t > 0) g += trans[prev * NN + lab] * mf;
        g += scores[(bT + t) * NN + lab] * mf;
        if (mi) lastidx = t;
        prev = lab;
    }
    g += endv[labels[bT + lastidx]];

    red[b] = logz[b] - g;
    __syncthreads();
    if (b == 0) {
        float acc = 0.f;
        for (int k = 0; k < BB; ++k) acc += red[k];
        loss_out[0] = acc / (float)BB;
    }
}

// ============================================================================
extern "C" void kernel_launch(void* const* d_in, const int* in_sizes, int n_in,
                              void* d_out, int out_size, void* d_ws, size_t ws_size,
                              hipStream_t stream) {
    const float* inputs = (const float*)d_in[0];   // (B,T,D)
    const int*   mask   = (const int*)  d_in[1];   // (B,T)
    const int*   labels = (const int*)  d_in[2];   // (B,T)
    const float* W      = (const float*)d_in[3];   // (D,N)
    const float* bias   = (const float*)d_in[4];   // (N)
    const float* trans  = (const float*)d_in[5];   // (N,N)
    const float* startv = (const float*)d_in[6];   // (N)
    const float* endv   = (const float*)d_in[7];   // (N)

    float* out    = (float*)d_out;
    float* scores = out;                            // B*T*N floats
    float* tags   = out + (size_t)MM * NN;          // B*T floats
    float* loss   = tags + MM;                      // 1 float

    // workspace layout
    float*         logz   = (float*)d_ws;                                // 256 B
    unsigned char* bp_buf = (unsigned char*)d_ws + 256;                  // 2 MB
    float2*        Wp     = (float2*)((unsigned char*)d_ws + 256 + (size_t)MM * NN);

    pack_w_kernel<<<((DD / 2) * NN + 255) / 256, 256, 0, stream>>>(W, Wp);
    crf_gemm_kernel<<<MM / 16, 128, 0, stream>>>(inputs, mask, Wp, bias, scores);
    crf_forward_kernel<<<BB, 256, 0, stream>>>(scores, mask, trans, startv, endv,
                                               logz, bp_buf, tags);
    crf_loss_kernel<<<1, BB, 0, stream>>>(scores, mask, labels, trans, startv,
                                          endv, logz, loss);
}